// FlashRecoModel_14826227105834
// MI455X (gfx1250) — compile-verified
//
#include <hip/hip_runtime.h>

typedef __attribute__((ext_vector_type(2))) float v2f;
typedef __attribute__((ext_vector_type(8))) float v8f;

#define B_ 8
#define F_ 128
#define P_ 256
#define T_ 2048

// ---------------------------------------------------------------------------
// Kernel 1: per-flash time-bin center + dense normalization (matches reference
// sum over ALL ticks). One thread per (b,f) -> 1024 threads total.
// ---------------------------------------------------------------------------
__global__ __launch_bounds__(256) void prep_norm_kernel(
    const float* __restrict__ ftime,   // [B,F,1]
    const int*   __restrict__ nt_p,    // scalar
    const float* __restrict__ sigma_p, // scalar
    float* __restrict__ tb_out,        // [B*F]
    float* __restrict__ inv_out)       // [B*F]
{
    int i = blockIdx.x * blockDim.x + threadIdx.x;
    if (i >= B_ * F_) return;
    const int   nt        = *nt_p;
    const float inv_sigma = 1.0f / (*sigma_p);
    float tb = ftime[i] * (float)nt;
    tb = fminf(fmaxf(tb, 0.0f), (float)(nt - 1));
    float s = 0.0f;
    for (int t = 0; t < nt; ++t) {
        float z = ((float)t - tb) * inv_sigma;
        s += __expf(-0.5f * z * z);
    }
    tb_out[i]  = tb;
    inv_out[i] = 1.0f / (s + 1e-10f);
}

// ---------------------------------------------------------------------------
// Kernel 2: A[b][p][f] = pe[b][f][p] * conf[b][f]   (transpose + scale, 1 MB)
// Thread mapping gives coalesced reads of pe.
// ---------------------------------------------------------------------------
__global__ __launch_bounds__(256) void prep_a_kernel(
    const float* __restrict__ pe,    // [B,F,P]
    const float* __restrict__ conf,  // [B,F,1]
    float* __restrict__ A)           // [B,P,F]
{
    int idx = blockIdx.x * blockDim.x + threadIdx.x;   // over B*F*P
    if (idx >= B_ * F_ * P_) return;
    int p  = idx % P_;
    int f  = (idx / P_) % F_;
    int b  = idx / (P_ * F_);
    float v = pe[idx] * conf[b * F_ + f];
    A[(b * P_ + p) * F_ + f] = v;
}

// ---------------------------------------------------------------------------
// Kernel 3: W[b][f][t] = exp(-0.5*((t-tb)/sigma)^2) * inv_norm   (8 MB)
// ---------------------------------------------------------------------------
__global__ __launch_bounds__(256) void prep_w_kernel(
    const float* __restrict__ tb_in,
    const float* __restrict__ inv_in,
    const float* __restrict__ sigma_p,
    float* __restrict__ W)           // [B,F,T]
{
    int idx = blockIdx.x * blockDim.x + threadIdx.x;   // over B*F*T
    if (idx >= B_ * F_ * T_) return;
    int t = idx % T_;
    int i = idx / T_;                 // (b*F + f)
    const float inv_sigma = 1.0f / (*sigma_p);
    float z = ((float)t - tb_in[i]) * inv_sigma;
    W[idx] = __expf(-0.5f * z * z) * inv_in[i];
}

// ---------------------------------------------------------------------------
// Kernel 4: batched GEMM via V_WMMA_F32_16X16X4_F32.
// One wave (32 lanes) computes one 16x16 tile of C_b = A_b[256x128] * W_b[128x2048].
// 8 waves per 256-thread block; 16384 tiles -> 2048 blocks.
// ---------------------------------------------------------------------------
__global__ __launch_bounds__(256) void gemm_wmma_kernel(
    const float* __restrict__ A,   // [B,P,F]
    const float* __restrict__ W,   // [B,F,T]
    float* __restrict__ out)       // [B,P,T]
{
    constexpr int tilesM = P_ / 16;          // 16
    constexpr int tilesN = T_ / 16;          // 128

    const int wave = threadIdx.x >> 5;       // 0..7
    const int lane = threadIdx.x & 31;
    const int half = lane >> 4;              // 0: lanes 0-15, 1: lanes 16-31
    const int sub  = lane & 15;

    const int tile = blockIdx.x * 8 + wave;  // 0 .. 16383 exactly
    const int b    = tile / (tilesM * tilesN);
    const int rem  = tile % (tilesM * tilesN);
    const int tm   = rem / tilesN;
    const int tn   = rem % tilesN;

    const float* __restrict__ Ab = A + (size_t)b * P_ * F_ + (tm * 16) * F_;
    const float* __restrict__ Wb = W + (size_t)b * F_ * T_ + tn * 16;

    v8f acc = {};
    // K loop: 128 / 4 = 32 WMMA ops
    #pragma unroll 4
    for (int k = 0; k < F_; k += 4) {
        // A fragment (16x4 f32, 2 VGPRs):
        //   lanes 0-15 : row=sub, K = k..k+1
        //   lanes 16-31: row=sub, K = k+2..k+3
        const float* ap = Ab + sub * F_ + k + 2 * half;
        v2f afrag;
        afrag.x = ap[0];
        afrag.y = ap[1];
        // B fragment (4x16 f32, 2 VGPRs):
        //   VGPR0: row K = k + 2*half, col = sub
        //   VGPR1: row K = k + 2*half + 1, col = sub
        const float* wp = Wb + (size_t)(k + 2 * half) * T_ + sub;
        v2f bfrag;
        bfrag.x = wp[0];
        bfrag.y = wp[T_];
        // 8 args: (neg_a, A, neg_b, B, c_mod, C, reuse_a, reuse_b)
        acc = __builtin_amdgcn_wmma_f32_16x16x4_f32(
            false, afrag, false, bfrag, (short)0, acc, false, false);
    }

    // C/D layout: VGPR r -> row = 8*half + r, col = sub
    float* op = out + (size_t)b * P_ * T_ + (size_t)(tm * 16 + 8 * half) * T_
              + tn * 16 + sub;
    #pragma unroll
    for (int r = 0; r < 8; ++r) {
        op[(size_t)r * T_] = acc[r];
    }
}

// ---------------------------------------------------------------------------
// Host-side launcher
// ---------------------------------------------------------------------------
extern "C" void kernel_launch(void* const* d_in, const int* in_sizes, int n_in,
                              void* d_out, int out_size, void* d_ws, size_t ws_size,
                              hipStream_t stream) {
    const float* pe     = (const float*)d_in[0];  // [B,F,P]
    const float* ftime  = (const float*)d_in[1];  // [B,F,1]
    const float* conf   = (const float*)d_in[2];  // [B,F,1]
    const int*   nt_p   = (const int*)d_in[3];    // scalar (2048)
    const float* sigma  = (const float*)d_in[4];  // scalar (1.0)

    float* out = (float*)d_out;                   // [B,P,T]

    // Workspace layout (floats): A [B*P*F] | W [B*F*T] | tb [B*F] | inv [B*F]
    float* wsA   = (float*)d_ws;
    float* wsW   = wsA + (size_t)B_ * P_ * F_;
    float* wsTb  = wsW + (size_t)B_ * F_ * T_;
    float* wsInv = wsTb + (size_t)B_ * F_;

    // 1) per-flash norm
    prep_norm_kernel<<<(B_ * F_ + 255) / 256, 256, 0, stream>>>(
        ftime, nt_p, sigma, wsTb, wsInv);
    // 2) scaled transposed A
    prep_a_kernel<<<(B_ * F_ * P_ + 255) / 256, 256, 0, stream>>>(
        pe, conf, wsA);
    // 3) dense normalized Gaussian weights
    prep_w_kernel<<<(B_ * F_ * T_ + 255) / 256, 256, 0, stream>>>(
        wsTb, wsInv, sigma, wsW);
    // 4) batched WMMA GEMM -> output
    const int totalTiles = B_ * (P_ / 16) * (T_ / 16);   // 16384
    gemm_wmma_kernel<<<totalTiles / 8, 256, 0, stream>>>(
        wsA, wsW, out);
}